// AlternateWeaveGather_14602888806816
// MI455X (gfx1250) — compile-verified
//
#include <hip/hip_runtime.h>
#include <hip/hip_bf16.h>

typedef __attribute__((ext_vector_type(2))) float v2f;
typedef __attribute__((ext_vector_type(8))) float v8f;

#define D_IN   512
#define D_OUT  512
#define NSEG   256
#define ROWS_PER_BLOCK 128

// ---------------------------------------------------------------------------
// Kernel 0: zero the workspace accumulators (segsum[256*512] + counts[256]).
// ---------------------------------------------------------------------------
__global__ void zero_kernel(float* __restrict__ p, int n) {
    int i = blockIdx.x * blockDim.x + threadIdx.x;
    if (i < n) p[i] = 0.0f;
}

// ---------------------------------------------------------------------------
// Kernel 1: segment-sum of x (sorted batch ids) -> segsum [NSEG, 512], counts.
// Block = 128 threads; thread t owns columns [4t, 4t+4) via float4 loads, so a
// block covers a full 512-float row. Each block walks ROWS_PER_BLOCK rows,
// accumulating in registers; flushes via f32 atomics only at segment changes
// (batch is sorted -> ~1-2 flushes per block).
// ---------------------------------------------------------------------------
__global__ __launch_bounds__(128)
void segsum_kernel(const float* __restrict__ x,
                   const int* __restrict__ batch,
                   float* __restrict__ segsum,
                   float* __restrict__ counts,
                   int n) {
    const int tid = threadIdx.x;
    const int col = tid * 4;
    int r0 = blockIdx.x * ROWS_PER_BLOCK;
    int r1 = r0 + ROWS_PER_BLOCK;
    if (r1 > n) r1 = n;
    if (r0 >= r1) return;

    float4 acc = make_float4(0.f, 0.f, 0.f, 0.f);
    int cur = batch[r0];
    int cnt = 0;

    #pragma unroll 4
    for (int r = r0; r < r1; ++r) {
        int s = batch[r];                 // uniform across the block
        if (s != cur) {                   // uniform branch
            float* dst = segsum + (size_t)cur * D_IN + col;
            atomicAdd(dst + 0, acc.x);
            atomicAdd(dst + 1, acc.y);
            atomicAdd(dst + 2, acc.z);
            atomicAdd(dst + 3, acc.w);
            if (tid == 0) atomicAdd(counts + cur, (float)cnt);
            acc = make_float4(0.f, 0.f, 0.f, 0.f);
            cnt = 0;
            cur = s;
        }
        const float4 v = *(const float4*)(x + (size_t)r * D_IN + col);
        acc.x += v.x; acc.y += v.y; acc.z += v.z; acc.w += v.w;
        ++cnt;
    }
    // final flush
    {
        float* dst = segsum + (size_t)cur * D_IN + col;
        atomicAdd(dst + 0, acc.x);
        atomicAdd(dst + 1, acc.y);
        atomicAdd(dst + 2, acc.z);
        atomicAdd(dst + 3, acc.w);
        if (tid == 0) atomicAdd(counts + cur, (float)cnt);
    }
}

// ---------------------------------------------------------------------------
// Kernel 2: in-place mean: segsum[s][d] /= max(counts[s], 1)
// ---------------------------------------------------------------------------
__global__ void mean_kernel(float* __restrict__ segsum,
                            const float* __restrict__ counts) {
    int idx = blockIdx.x * blockDim.x + threadIdx.x;   // NSEG*D_IN total
    int s = idx >> 9;                                  // /512
    float c = counts[s];
    segsum[idx] = segsum[idx] / fmaxf(c, 1.0f);
}

// ---------------------------------------------------------------------------
// Kernel 3: out[M=256, N=512] = mean[256,512] @ W^T[512,512] + b, via
// V_WMMA_F32_16X16X4_F32. One wave per 16x16 output tile; K=512 in steps of 4.
//
// ISA VGPR layouts (CDNA5 ISA):
//   A 16x4 f32:  lane l -> row m = l&15; lanes 0-15 hold K=0,1 in v0,v1;
//                lanes 16-31 hold K=2,3.  => a = A[m][k + 2*(l>>4) + {0,1}]
//   B 4x16 f32:  VGPR v -> row K = v + 2*(l>>4); col n = l&15.
//                B[k][n] = W^T[k][n] = W[n][k]  (contiguous float2 in a W row)
//   C/D 16x16:   VGPR v -> row = v + 8*(l>>4); col = l&15.
// Grid exactly covers all tiles -> EXEC is all ones around every WMMA.
// ---------------------------------------------------------------------------
__global__ __launch_bounds__(256)
void gemm_wmma_kernel(const float* __restrict__ mean,   // [256, 512]
                      const float* __restrict__ W,      // [512, 512] (row = out)
                      const float* __restrict__ bias,   // [512]
                      float* __restrict__ out) {        // [256, 512]
    const int wave = threadIdx.x >> 5;                  // 8 waves / block
    const int lane = threadIdx.x & 31;
    const int half = lane >> 4;                         // 0 or 1
    const int ml   = lane & 15;

    const int tileId = blockIdx.x * 8 + wave;           // 512 tiles total
    const int tr = tileId >> 5;                         // 16 row tiles (M)
    const int tc = tileId & 31;                         // 32 col tiles (N)

    const float* Arow = mean + (size_t)(tr * 16 + ml) * D_IN;  // A row m
    const float* Wrow = W    + (size_t)(tc * 16 + ml) * D_IN;  // B col n = W row n

    v8f acc = {};
    #pragma unroll 8
    for (int k = 0; k < D_IN; k += 4) {
        const int kk = k + 2 * half;
        const v2f a = *(const v2f*)(Arow + kk);   // A[m][kk], A[m][kk+1]
        const v2f b = *(const v2f*)(Wrow + kk);   // B[kk][n], B[kk+1][n]
        acc = __builtin_amdgcn_wmma_f32_16x16x4_f32(
            /*neg_a=*/false, a, /*neg_b=*/false, b,
            /*c_mod=*/(short)0, acc, /*reuse_a=*/false, /*reuse_b=*/false);
    }

    const int   ncol = tc * 16 + ml;
    const float bv   = bias[ncol];
    #pragma unroll
    for (int v = 0; v < 8; ++v) {
        const int row = tr * 16 + v + 8 * half;
        out[(size_t)row * D_OUT + ncol] = acc[v] + bv;
    }
}

// ---------------------------------------------------------------------------
// Launch: inputs in setup_inputs() order: x, W, b, batch, num_segments
// ---------------------------------------------------------------------------
extern "C" void kernel_launch(void* const* d_in, const int* in_sizes, int n_in,
                              void* d_out, int out_size, void* d_ws, size_t ws_size,
                              hipStream_t stream) {
    const float* x     = (const float*)d_in[0];
    const float* W     = (const float*)d_in[1];
    const float* bias  = (const float*)d_in[2];
    const int*   batch = (const int*)d_in[3];

    const int n = in_sizes[0] / D_IN;         // 131072

    float* segsum = (float*)d_ws;             // [NSEG * D_IN]
    float* counts = segsum + NSEG * D_IN;     // [NSEG]
    float* out    = (float*)d_out;            // [NSEG * D_OUT]

    // 0) zero accumulators (must be re-zeroed every call: deterministic)
    {
        const int nz = NSEG * D_IN + NSEG;
        zero_kernel<<<(nz + 255) / 256, 256, 0, stream>>>(segsum, nz);
    }

    // 1) segment-sum (bandwidth-bound pass over x)
    {
        const int nblocks = (n + ROWS_PER_BLOCK - 1) / ROWS_PER_BLOCK;  // 1024
        segsum_kernel<<<nblocks, 128, 0, stream>>>(x, batch, segsum, counts, n);
    }

    // 2) divide by counts -> segment means (in place)
    mean_kernel<<<(NSEG * D_IN) / 256, 256, 0, stream>>>(segsum, counts);

    // 3) tiny f32 WMMA GEMM: out = mean @ W^T + b
    //    512 tiles, 8 waves (256 thr) per block -> 64 blocks
    gemm_wmma_kernel<<<64, 256, 0, stream>>>(segsum, W, bias, out);
}